// GAT_32117765439616
// MI455X (gfx1250) — compile-verified
//
#include <hip/hip_runtime.h>

typedef __attribute__((ext_vector_type(2))) float v2f;
typedef __attribute__((ext_vector_type(8))) float v8f;
typedef __attribute__((ext_vector_type(4))) unsigned int u32x4;
typedef __attribute__((ext_vector_type(4))) int i32x4;
typedef __attribute__((ext_vector_type(8))) int i32x8;

#define NEG_SLOPE 0.2f

// ---- Stage W[K x 64] (f32, row-major in global) into LDS COLUMN-MAJOR sW[n*K + k] ----
// Uses the Tensor Data Mover as a 3-D transpose gather when available.
template<int K>
__device__ __forceinline__ void stage_W(const float* __restrict__ W, float* sW)
{
#if __has_builtin(__builtin_amdgcn_tensor_load_to_lds) && __has_builtin(__builtin_amdgcn_s_wait_tensorcnt)
  if (threadIdx.x < 32) {   // one wave issues the TDM descriptor (EXEC ignored by TDM)
    // flat LDS aperture: addr[31:0] is the LDS byte offset (ISA 10.2)
    unsigned lds_addr = (unsigned)(unsigned long long)sW;
    unsigned long long ga = (unsigned long long)W;
    u32x4 g0;
    g0.x = 1u;                                    // count=1, is_restore=0, gather=0
    g0.y = lds_addr;                              // lds_addr
    g0.z = (unsigned)ga;                          // global_addr[31:0]
    g0.w = (unsigned)(ga >> 32) | 0x80000000u;    // global_addr[56:32] | type=2
    // 3-D transpose: X: 1 elem; Y: k (stride 64 elems); Z: n (stride 1 elem)
    // => reads element (k*64+n), writes LDS contiguously as [n][k]
    i32x8 g1;
    g1[0] = 0x20000;                              // data_size=2 (4 bytes); mask=0
    g1[1] = (1 << 16);                            // tensor_dim0 = 1 (lo16)
    g1[2] = (K << 16);                            // dim0 hi=0 | tensor_dim1 lo16 = K
    g1[3] = (1 << 16);                            // dim1 hi=0 | tile_dim0 = 1
    g1[4] = K | (64 << 16);                       // tile_dim1 = K | tile_dim2 = 64
    g1[5] = 64;                                   // tensor_dim0_stride lo32 = 64
    g1[6] = (1 << 16);                            // dim0_stride hi=0 | tensor_dim1_stride lo16 = 1
    g1[7] = 0;                                    // dim1_stride hi = 0
    i32x4 g2 = {64, 0, 0, 0};                     // tensor_dim2 = 64; dim2_stride unused
    i32x4 g3 = {0, 0, 0, 0};
#if defined(__clang_major__) && __clang_major__ >= 23
    i32x8 z8 = {0, 0, 0, 0, 0, 0, 0, 0};
    __builtin_amdgcn_tensor_load_to_lds(g0, g1, g2, g3, z8, 0);
#else
    __builtin_amdgcn_tensor_load_to_lds(g0, g1, g2, g3, 0);
#endif
  }
  __builtin_amdgcn_s_wait_tensorcnt(0);           // nop for non-issuing waves
#else
  for (int i = threadIdx.x; i < K * 64; i += 256) // i = n*K + k; consecutive k -> no bank conflicts
    sW[i] = W[(i % K) * 64 + (i / K)];
#endif
  __syncthreads();
}

// ---------------- WMMA GEMM: C[nrows x 64] = A[nrows x K] @ W[K x 64] ---------------
template<int K>
__global__ __launch_bounds__(256) void gemm_wmma_kernel(
    const float* __restrict__ A, const float* __restrict__ W,
    float* __restrict__ C, int nrows)
{
  __shared__ float sW[K * 64];                 // column-major: sW[n*K + k]
  stage_W<K>(W, sW);

  const int wave = threadIdx.x >> 5;
  const int lane = threadIdx.x & 31;
  const int m0   = (blockIdx.x * 8 + wave) * 16;
  const int row  = lane & 15;
  const int kh   = (lane >> 4) << 1;           // 0 or 2: K-pair selector (ISA 7.12.2)
  int gr = m0 + row;
  gr = gr < nrows ? gr : (nrows - 1);          // clamp: keeps EXEC all-ones, no branches
  const float* arow = A + (size_t)gr * K;

  const int nl = lane & 15;
  // one LDS base; the 4 column tiles are immediate offsets (16*K, 32*K, 48*K dwords)
  const float* wbase = sW + nl * K + kh;

  v8f acc0 = {}, acc1 = {}, acc2 = {}, acc3 = {};
#pragma unroll
  for (int k = 0; k < K; k += 4) {
    v2f a;
    a.x = arow[k + kh];
    a.y = arow[k + kh + 1];
    v2f b0 = *(const v2f*)(wbase + k);           // {W[k+kh][nl],   W[k+kh+1][nl]}
    v2f b1 = *(const v2f*)(wbase + k + 16 * K);  //  ... column nl+16
    v2f b2 = *(const v2f*)(wbase + k + 32 * K);  //  ... column nl+32
    v2f b3 = *(const v2f*)(wbase + k + 48 * K);  //  ... column nl+48
    acc0 = __builtin_amdgcn_wmma_f32_16x16x4_f32(false, a, false, b0, (short)0, acc0, false, false);
    acc1 = __builtin_amdgcn_wmma_f32_16x16x4_f32(false, a, false, b1, (short)0, acc1, false, false);
    acc2 = __builtin_amdgcn_wmma_f32_16x16x4_f32(false, a, false, b2, (short)0, acc2, false, false);
    acc3 = __builtin_amdgcn_wmma_f32_16x16x4_f32(false, a, false, b3, (short)0, acc3, false, false);
  }

  const int rbase = m0 + ((lane >> 4) << 3);   // C/D layout: M = v + 8*(lane/16)
  const int ncol  = lane & 15;
#pragma unroll
  for (int v = 0; v < 8; ++v) {
    const int r = rbase + v;
    if (r < nrows) {                           // uniform per 16-lane half
      float* crow = C + (size_t)r * 64 + ncol;
      crow[0]  = acc0[v];
      crow[16] = acc1[v];
      crow[32] = acc2[v];
      crow[48] = acc3[v];
    }
  }
}

// ---------------- per-node attention logits ----------------
__global__ void att_kernel(const float* __restrict__ h,
                           const float* __restrict__ a_src,
                           const float* __restrict__ a_dst,
                           float* __restrict__ alpha_s,
                           float* __restrict__ alpha_d, int n)
{
  int i = blockIdx.x * blockDim.x + threadIdx.x;
  if (i >= n) return;
  const float* hr = h + (size_t)i * 64;
  float s = 0.f, d = 0.f;
#pragma unroll
  for (int f = 0; f < 64; ++f) { float v = hr[f]; s += v * a_src[f]; d += v * a_dst[f]; }
  alpha_s[i] = s;
  alpha_d[i] = d;
}

// monotonic int mapping for float atomic max
__device__ __forceinline__ int fmap(float f) {
  int b = __float_as_int(f);
  return b >= 0 ? b : (b ^ 0x7fffffff);
}
__device__ __forceinline__ float funmap(int b) {
  return __int_as_float(b >= 0 ? b : (b ^ 0x7fffffff));
}

// ---------------- per-layer init: acc=0, denom=0, emax=-inf, stats=0 ----------------
__global__ void init_layer_kernel(float* __restrict__ acc, float* __restrict__ denom,
                                  int* __restrict__ emax, float* __restrict__ stats, int n)
{
  int i = blockIdx.x * blockDim.x + threadIdx.x;
  if (i < n * 64) acc[i] = 0.f;
  if (i < n) { denom[i] = 0.f; emax[i] = (int)0x807FFFFF; }  // fmap(-inf)
  if (i < 128) stats[i] = 0.f;
}

__device__ __forceinline__ void edge_ids(const int* __restrict__ src,
                                         const int* __restrict__ dst,
                                         int e, int E, int& s, int& d) {
  if (e < E) { s = src[e]; d = dst[e]; } else { s = d = e - E; }
}

// ---------------- pass 1: segment max of leaky-relu logits ----------------
__global__ void edge_max_kernel(const int* __restrict__ src, const int* __restrict__ dst,
                                const float* __restrict__ as_, const float* __restrict__ ad_,
                                int* __restrict__ emax, int E, int n)
{
  int e = blockIdx.x * blockDim.x + threadIdx.x;
  if (e >= E + n) return;
  int s, d; edge_ids(src, dst, e, E, s, d);
  float x = as_[s] + ad_[d];
  x = x > 0.f ? x : NEG_SLOPE * x;
  atomicMax(&emax[d], fmap(x));
}

// ---------------- pass 2: exp(x - max) and segment sum ----------------
__global__ void edge_exp_kernel(const int* __restrict__ src, const int* __restrict__ dst,
                                const float* __restrict__ as_, const float* __restrict__ ad_,
                                const int* __restrict__ emax, float* __restrict__ eexp,
                                float* __restrict__ denom, int E, int n)
{
  int e = blockIdx.x * blockDim.x + threadIdx.x;
  if (e >= E + n) return;
  int s, d; edge_ids(src, dst, e, E, s, d);
  float x = as_[s] + ad_[d];
  x = x > 0.f ? x : NEG_SLOPE * x;
  float ex = __expf(x - funmap(emax[d]));
  eexp[e] = ex;
  atomicAdd(&denom[d], ex);
}

// -------- pass 3: acc[dst] += eexp * h[src]  (16 threads/edge, float4 gather) -------
__global__ void edge_aggr_kernel(const int* __restrict__ src, const int* __restrict__ dst,
                                 const float* __restrict__ eexp, const float* __restrict__ h,
                                 float* __restrict__ acc, int E, int n)
{
  int idx = blockIdx.x * blockDim.x + threadIdx.x;
  int e = idx >> 4;
  if (e >= E + n) return;
  int q = (idx & 15) << 2;
  int s, d; edge_ids(src, dst, e, E, s, d);
  float w = eexp[e];
  float4 v = *(const float4*)(h + (size_t)s * 64 + q);
  float* a = acc + (size_t)d * 64 + q;
  atomicAdd(a + 0, w * v.x);
  atomicAdd(a + 1, w * v.y);
  atomicAdd(a + 2, w * v.z);
  atomicAdd(a + 3, w * v.w);
}

// ---------------- finalize: normalize by denom, add bias ----------------
__global__ void finalize_kernel(float* __restrict__ acc, const float* __restrict__ denom,
                                const float* __restrict__ bias, int n)
{
  int i = blockIdx.x * blockDim.x + threadIdx.x;
  if (i >= n * 64) return;
  int node = i >> 6, f = i & 63;
  acc[i] = acc[i] / (denom[node] + 1e-16f) + bias[f];
}

// ---------------- batchnorm: per-feature sum / sumsq ----------------
__global__ __launch_bounds__(256) void bn_stats_kernel(const float* __restrict__ acc,
                                                       float* __restrict__ stats, int n)
{
  __shared__ float ssum[64], ssq[64];
  int t = threadIdx.x;
  if (t < 64) { ssum[t] = 0.f; ssq[t] = 0.f; }
  __syncthreads();
  int f = t & 63;
  float s = 0.f, q = 0.f;
  for (int r = blockIdx.x * 4 + (t >> 6); r < n; r += gridDim.x * 4) {
    float v = acc[(size_t)r * 64 + f];
    s += v; q += v * v;
  }
  atomicAdd(&ssum[f], s);
  atomicAdd(&ssq[f], q);
  __syncthreads();
  if (t < 64) { atomicAdd(&stats[t], ssum[t]); atomicAdd(&stats[64 + t], ssq[t]); }
}

__global__ void bn_apply_kernel(const float* __restrict__ acc, const float* __restrict__ stats,
                                const float* __restrict__ gamma, const float* __restrict__ beta,
                                float* __restrict__ h, int n)
{
  int i = blockIdx.x * blockDim.x + threadIdx.x;
  if (i >= n * 64) return;
  int f = i & 63;
  float invN = 1.0f / (float)n;
  float mean = stats[f] * invN;
  float var  = stats[64 + f] * invN - mean * mean;
  float v = (acc[i] - mean) * rsqrtf(var + 1e-5f) * gamma[f] + beta[f];
  h[i] = v > 0.f ? v : 0.f;
}

extern "C" void kernel_launch(void* const* d_in, const int* in_sizes, int n_in,
                              void* d_out, int out_size, void* d_ws, size_t ws_size,
                              hipStream_t stream) {
  const float* x        = (const float*)d_in[0];
  const int*   ei       = (const int*)d_in[1];
  const float* W1       = (const float*)d_in[2];
  const float* att_src1 = (const float*)d_in[3];
  const float* att_dst1 = (const float*)d_in[4];
  const float* bias1    = (const float*)d_in[5];
  const float* gamma    = (const float*)d_in[6];
  const float* beta     = (const float*)d_in[7];
  const float* W2       = (const float*)d_in[8];
  const float* att_src2 = (const float*)d_in[9];
  const float* att_dst2 = (const float*)d_in[10];
  const float* bias2    = (const float*)d_in[11];

  const int N = in_sizes[0] / 128;   // 50000
  const int E = in_sizes[1] / 2;     // 800000
  const int EN = E + N;              // with self loops

  const int* src = ei;
  const int* dst = ei + E;

  // workspace layout (fp32; ~30 MB, fully L2-resident on MI455X)
  float* ws    = (float*)d_ws;
  float* h     = ws;                         // N*64
  float* acc   = h + (size_t)N * 64;         // N*64
  float* as_   = acc + (size_t)N * 64;       // N
  float* ad_   = as_ + N;                    // N
  float* denom = ad_ + N;                    // N
  float* eexp  = denom + N;                  // E+N
  int*   emax  = (int*)(eexp + EN);          // N
  float* stats = (float*)(emax + N);         // 128

  float* out = (float*)d_out;

  const int T = 256;
  dim3 blk(T);
  dim3 gN64((N * 64 + T - 1) / T);
  dim3 gN((N + T - 1) / T);
  dim3 gE((EN + T - 1) / T);
  dim3 gE16(((size_t)EN * 16 + T - 1) / T);
  dim3 gGemm((N + 127) / 128);               // 8 waves x 16 rows per block

  // ---------------- layer 1 ----------------
  gemm_wmma_kernel<128><<<gGemm, blk, 0, stream>>>(x, W1, h, N);
  att_kernel<<<gN, blk, 0, stream>>>(h, att_src1, att_dst1, as_, ad_, N);
  init_layer_kernel<<<gN64, blk, 0, stream>>>(acc, denom, emax, stats, N);
  edge_max_kernel<<<gE, blk, 0, stream>>>(src, dst, as_, ad_, emax, E, N);
  edge_exp_kernel<<<gE, blk, 0, stream>>>(src, dst, as_, ad_, emax, eexp, denom, E, N);
  edge_aggr_kernel<<<gE16, blk, 0, stream>>>(src, dst, eexp, h, acc, E, N);
  finalize_kernel<<<gN64, blk, 0, stream>>>(acc, denom, bias1, N);

  // ---------------- batchnorm + relu ----------------
  bn_stats_kernel<<<dim3(256), blk, 0, stream>>>(acc, stats, N);
  bn_apply_kernel<<<gN64, blk, 0, stream>>>(acc, stats, gamma, beta, h, N);

  // ---------------- layer 2 ----------------
  gemm_wmma_kernel<64><<<gGemm, blk, 0, stream>>>(h, W2, acc, N);
  att_kernel<<<gN, blk, 0, stream>>>(acc, att_src2, att_dst2, as_, ad_, N);
  init_layer_kernel<<<gN64, blk, 0, stream>>>(out, denom, emax, stats, N);
  edge_max_kernel<<<gE, blk, 0, stream>>>(src, dst, as_, ad_, emax, E, N);
  edge_exp_kernel<<<gE, blk, 0, stream>>>(src, dst, as_, ad_, emax, eexp, denom, E, N);
  edge_aggr_kernel<<<gE16, blk, 0, stream>>>(src, dst, eexp, acc, out, E, N);
  finalize_kernel<<<gN64, blk, 0, stream>>>(out, denom, bias2, N);
}